// LC_OPE_25950192402885
// MI455X (gfx1250) — compile-verified
//
#include <hip/hip_runtime.h>
#include <math.h>

#ifndef __has_builtin
#define __has_builtin(x) 0
#endif

#if __has_builtin(__builtin_amdgcn_global_load_async_to_lds_b128)
#define ASYNC_LOAD 1
#else
#define ASYNC_LOAD 0
#endif

#if __has_builtin(__builtin_amdgcn_global_store_async_from_lds_b128)
#define ASYNC_STORE 1
#else
#define ASYNC_STORE 0
#endif

#define ROWW 149                        // 3*49 + 2 floats per row
#define TILE_ROWS 64
#define TILE_FLOATS (TILE_ROWS * ROWW)  // 9536 floats = 38144 B (16B aligned)
#define TILE_V4 (TILE_FLOATS / 4)       // 2384 16-byte chunks
#define OUT_FLOATS (TILE_ROWS * 3)      // 192 floats = 48 16-byte chunks
#define OUT_V4 (OUT_FLOATS / 4)         // 48
#define BLOCK 64

typedef __attribute__((ext_vector_type(4))) float f4;
typedef __attribute__((ext_vector_type(4))) int   i4;
typedef __attribute__((address_space(1))) i4 gi4;   // global int4
typedef __attribute__((address_space(3))) i4 li4;   // LDS int4

__device__ __forceinline__ void wait_asynccnt0() {
#if __has_builtin(__builtin_amdgcn_s_wait_asynccnt)
  __builtin_amdgcn_s_wait_asynccnt(0);
#else
  asm volatile("s_wait_asynccnt 0" ::: "memory");
#endif
}

__device__ __forceinline__ void embed7(float v, float e[7]) {
  const float OMEGA = 1.57079632679489662f;  // pi/2
  const float R2    = 1.41421356237309505f;  // sqrt(2)
  float t  = v * OMEGA;
  float s1 = sinf(t);
  float c1 = cosf(t);
  float s2 = 2.0f * s1 * c1;
  float c2 = fmaf(2.0f * c1, c1, -1.0f);
  float s3 = fmaf(s2, c1,  c2 * s1);
  float c3 = fmaf(c2, c1, -s2 * s1);
  e[0] = 1.0f;
  e[1] = R2 * c1; e[2] = R2 * s1;
  e[3] = R2 * c2; e[4] = R2 * s2;
  e[5] = R2 * c3; e[6] = R2 * s3;
}

__global__ __launch_bounds__(BLOCK)
void lc_ope_kernel(const float* __restrict__ x, float* __restrict__ out, int nrows) {
  __shared__ float tile[TILE_FLOATS];
  __shared__ float outs[OUT_FLOATS];
  const int tid = threadIdx.x;
  const long long rowBase = (long long)blockIdx.x * TILE_ROWS;
  const float* gtile = x + rowBase * (long long)ROWW;
  const long long remRows = (long long)nrows - rowBase;
  const int validRows = (remRows >= TILE_ROWS) ? TILE_ROWS : (int)remRows;
  const bool fullTile = (validRows == TILE_ROWS);

  // ---- Stage input tile: global -> LDS, fully coalesced ----
  if (fullTile) {
#if ASYNC_LOAD
    // CDNA5 async copy: bypasses VGPRs, ASYNCcnt-tracked, 512B per instruction.
    gi4* g4 = (gi4*)gtile;
    li4* l4 = (li4*)tile;
    for (int i = tid; i < TILE_V4; i += BLOCK) {
      __builtin_amdgcn_global_load_async_to_lds_b128(g4 + i, l4 + i, 0, 0);
    }
    wait_asynccnt0();
#else
    const f4* g4 = (const f4*)gtile;
    f4* l4 = (f4*)tile;
    for (int i = tid; i < TILE_V4; i += BLOCK) l4[i] = g4[i];
#endif
  } else if (validRows > 0) {
    const int nf = validRows * ROWW;
    for (int i = tid; i < nf; i += BLOCK) tile[i] = gtile[i];
  }
  __syncthreads();

  // ---- Per-row bilinear form out[k] = X^T * L_k * Y ----
  if (tid < validRows) {
    const float* row = &tile[tid * ROWW];  // stride 149 dwords: odd -> conflict-free
    float X[7], Y[7];
    embed7(row[147], X);
    embed7(row[148], Y);
    float a0 = 0.0f, a1 = 0.0f, a2 = 0.0f;
#pragma unroll
    for (int i = 0; i < 7; ++i) {
#pragma unroll
      for (int j = 0; j < 7; ++j) {
        const float w = X[i] * Y[j];
        const int c = i * 7 + j;
        a0 = fmaf(w, row[c],      a0);
        a1 = fmaf(w, row[49 + c], a1);
        a2 = fmaf(w, row[98 + c], a2);
      }
    }
    if (fullTile) {
      // stride-3 writes: gcd(3,64)=1 -> conflict-free
      outs[tid * 3 + 0] = a0 * 0.01f;
      outs[tid * 3 + 1] = a1 * 0.01f;
      outs[tid * 3 + 2] = a2 * 0.01f;
    } else {
      const long long r = rowBase + tid;
      out[r * 3 + 0] = a0 * 0.01f;
      out[r * 3 + 1] = a1 * 0.01f;
      out[r * 3 + 2] = a2 * 0.01f;
    }
  }

  // ---- Coalesced output: LDS -> global (768B tile, 768B-aligned base) ----
  if (fullTile) {
    __syncthreads();
#if ASYNC_STORE
    gi4* o4 = (gi4*)(out + rowBase * 3);
    li4* s4 = (li4*)outs;
    if (tid < OUT_V4) {
      __builtin_amdgcn_global_store_async_from_lds_b128(o4 + tid, s4 + tid, 0, 0);
    }
    wait_asynccnt0();   // (S_ENDPGM also implicitly waits idle)
#else
    f4* o4 = (f4*)(out + rowBase * 3);
    const f4* s4 = (const f4*)outs;
    if (tid < OUT_V4) o4[tid] = s4[tid];
#endif
  }
}

extern "C" void kernel_launch(void* const* d_in, const int* in_sizes, int n_in,
                              void* d_out, int out_size, void* d_ws, size_t ws_size,
                              hipStream_t stream) {
  (void)n_in; (void)out_size; (void)d_ws; (void)ws_size;
  const float* x = (const float*)d_in[0];
  float* out = (float*)d_out;
  const int nrows = in_sizes[0] / ROWW;                     // 2,000,000
  const int nblocks = (nrows + TILE_ROWS - 1) / TILE_ROWS;  // 31,250
  lc_ope_kernel<<<nblocks, BLOCK, 0, stream>>>(x, out, nrows);
}